// DifferentiableTMO_10187662426935
// MI455X (gfx1250) — compile-verified
//
#include <hip/hip_runtime.h>

// ---------------------------------------------------------------------------
// DifferentiableTMO for MI455X (gfx1250, wave32)
//
// Stage 0 (per block): stage zero-padded WMMA operands into LDS
//   sHb[256][28] = H_basis padded (cols 25..27 = 0), sW[16][28] = weights
//   padded (rows 8..15 and cols 25..27 = 0).  All subsequent WMMA operand
//   fetches are unconditional ds_load_b64 -> no exec-mask branching.
// Stage A (per block, WMMA): CRF[k] = f0[k] + sum_n H_basis[k,n] * w[b,n]
//   via V_WMMA_F32_16X16X4_F32 (M = 16 CRF samples, N = batch 8-of-16,
//   K = basis dim, 7 chunks of 4).
// Stage B (per block): LDS acceleration structure:
//   - seg[j]  = (E[j], E[j+1], c[j], (c[j+1]-c[j])/(E[j+1]-E[j]))  (255 x b128)
//   - grid[c] = last j with E[j] <= c/G   (2048 x u8, exact scan seed)
// Stage C (streaming, HBM roofline ~17us for 398 MB @ 23.3 TB/s):
//   nontemporal b128 load -> clamp -> grid seed -> short segment scan ->
//   fma -> clip -> nontemporal b128 store; global_prefetch one iteration ahead.
// ---------------------------------------------------------------------------

typedef float v2f __attribute__((ext_vector_type(2)));
typedef float v4f __attribute__((ext_vector_type(4)));
typedef float v8f __attribute__((ext_vector_type(8)));

#define KS       256                  // CRF sample count
#define NBASIS   25                   // PCA basis count
#define NBPAD    28                   // basis dim padded to 7 WMMA K-chunks
#define GCELLS   2048                 // uniform grid cells (u8 entries)
#define BLOCK    256                  // 8 waves (wave32)
#define GX       160                  // blocks per batch
#define NPIX_B   (3 * 1080 * 1920)    // pixels per batch item = 6220800
#define N4_B     (NPIX_B / 4)         // float4 per batch item = 1555200

__global__ __launch_bounds__(BLOCK) void tmo_interp_kernel(
    const float* __restrict__ hdr,   // [8, NPIX_B]
    const float* __restrict__ w,     // [8, 25]
    const float* __restrict__ E,     // [256] strictly increasing
    const float* __restrict__ f0,    // [256]
    const float* __restrict__ Hb,    // [256, 25]
    float* __restrict__ out)         // [8, NPIX_B]
{
    __shared__ float         sE[KS];
    __shared__ float         sC[KS];
    __shared__ v4f           sSeg[KS];        // 255 used
    __shared__ unsigned char sGrid[GCELLS];
    __shared__ float         sHb[KS][NBPAD];  // zero-padded basis
    __shared__ float         sW[16][NBPAD];   // zero-padded weights

    const int tid  = threadIdx.x;
    const int b    = blockIdx.y;              // batch index 0..7
    const int wave = tid >> 5;
    const int lane = tid & 31;
    const int lh   = lane & 15;               // lane % 16
    const int khi  = (lane >> 4) << 1;        // 0 (lanes 0-15) or 2 (16-31)

    // ---- Stage 0: stage E + zero-padded WMMA operands into LDS ---------
    sE[tid] = E[tid];                          // BLOCK == KS
    for (int i = tid; i < KS * NBASIS; i += BLOCK)         // 25 iters, coalesced
        sHb[i / NBASIS][i % NBASIS] = Hb[i];
    for (int i = tid; i < KS * (NBPAD - NBASIS); i += BLOCK) // pad cols -> 0
        sHb[i / (NBPAD - NBASIS)][NBASIS + i % (NBPAD - NBASIS)] = 0.0f;
    for (int i = tid; i < 16 * NBPAD; i += BLOCK) {        // 2 iters
        const int r = i / NBPAD, c = i % NBPAD;
        (&sW[0][0])[i] = (r < 8 && c < NBASIS) ? w[r * NBASIS + c] : 0.0f;
    }
    __syncthreads();

    // ---- Stage A: CRF curve via f32 WMMA (branchless operand fetch) ----
    // A layout: lane = M (lh), VGPR pair = K {khi, khi+1}.
    // B layout: lane = N (lh), VGPR pair = K {khi, khi+1}.
    for (int t = 0; t < 2; ++t) {
        const int tile = wave * 2 + t;
        const int m    = tile * 16 + lh;       // A-matrix row (CRF sample)
        v8f acc = {0.f, 0.f, 0.f, 0.f, 0.f, 0.f, 0.f, 0.f};
        #pragma unroll
        for (int c = 0; c < 7; ++c) {
            const int kk = c * 4 + khi;        // even -> 8B-aligned ds_load_b64
            const v2f a  = *(const v2f*)&sHb[m][kk];
            const v2f bb = *(const v2f*)&sW[lh][kk];
            acc = __builtin_amdgcn_wmma_f32_16x16x4_f32(
                      false, a, false, bb, (short)0, acc, false, false);
        }
        // D layout: acc[r], lane l -> D[M = r + (l<16?0:8)][N = l%16].
        if (lh == b) {
            #pragma unroll
            for (int r = 0; r < 8; ++r) {
                const int k = tile * 16 + r + ((lane < 16) ? 0 : 8);
                sC[k] = f0[k] + acc[r];
            }
        }
    }
    __syncthreads();

    // ---- Stage B: segment table + uniform-grid seed --------------------
    for (int k = tid; k < KS - 1; k += BLOCK) {
        const float e0 = sE[k], e1 = sE[k + 1];
        const float c0 = sC[k], c1 = sC[k + 1];
        v4f s;
        s.x = e0;
        s.y = e1;
        s.z = c0;
        s.w = (c1 - c0) / (e1 - e0);           // E strictly increasing: safe
        sSeg[k] = s;
    }
    for (int cell = tid; cell < GCELLS; cell += BLOCK) {
        const float tleft = (float)cell * (1.0f / (float)GCELLS);
        int lo = 0, hi = KS;                    // upper_bound over sE
        while (lo < hi) {
            const int mid = (lo + hi) >> 1;
            if (sE[mid] <= tleft) lo = mid + 1; else hi = mid;
        }
        int j = lo - 1;
        j = (j < 0) ? 0 : ((j > KS - 2) ? KS - 2 : j);
        sGrid[cell] = (unsigned char)j;
    }
    __syncthreads();

    // ---- Stage C: streaming interp (HBM-bound) -------------------------
    const float E0   = sE[0];
    const float Emax = sE[KS - 1];
    const v4f* __restrict__ in4 =
        (const v4f*)(hdr + (size_t)b * (size_t)NPIX_B);
    v4f* __restrict__ out4 = (v4f*)(out + (size_t)b * (size_t)NPIX_B);

    for (int i = blockIdx.x * BLOCK + tid; i < N4_B; i += GX * BLOCK) {
        // Pull next iteration's line into L2 (speculative, safe past end).
        __builtin_prefetch(&in4[i + GX * BLOCK], 0, 1);
        const v4f v = __builtin_nontemporal_load(&in4[i]);
        v4f r;
        #pragma unroll
        for (int c = 0; c < 4; ++c) {
            float x = v[c];
            x = (x < E0) ? E0 : ((x > Emax) ? Emax : x);   // np.interp clamp
            int cell = (int)(x * (float)GCELLS);
            cell = (cell > GCELLS - 1) ? GCELLS - 1 : cell;
            int j = (int)sGrid[cell];
            v4f s = sSeg[j];
            while (x >= s.y && j < KS - 2) { ++j; s = sSeg[j]; }
            float y = fmaf(x - s.x, s.w, s.z);
            y = (y < 0.0f) ? 0.0f : ((y > 1.0f) ? 1.0f : y); // final clip
            r[c] = y;
        }
        __builtin_nontemporal_store(r, &out4[i]);
    }
}

extern "C" void kernel_launch(void* const* d_in, const int* in_sizes, int n_in,
                              void* d_out, int out_size, void* d_ws, size_t ws_size,
                              hipStream_t stream) {
    (void)in_sizes; (void)n_in; (void)d_ws; (void)ws_size; (void)out_size;
    const float* hdr = (const float*)d_in[0];   // [8,3,1080,1920]
    const float* w   = (const float*)d_in[1];   // [8,25]
    const float* E   = (const float*)d_in[2];   // [256]
    const float* f0  = (const float*)d_in[3];   // [256]
    const float* Hb  = (const float*)d_in[4];   // [256,25]
    float* out = (float*)d_out;

    dim3 grid(GX, 8);
    tmo_interp_kernel<<<grid, dim3(BLOCK), 0, stream>>>(hdr, w, E, f0, Hb, out);
}